// ESRNN_90005334655848
// MI455X (gfx1250) — compile-verified
//
#include <hip/hip_runtime.h>

#define T_CONST 720
#define B_CONST 50000
#define S_CONST 7
#define NT 718   // number of mean_sq_log_diff entries (T-2)

typedef __attribute__((ext_vector_type(2))) float v2f;
typedef __attribute__((ext_vector_type(8))) float v8f;

// One 7-position step of the ES-RNN recurrence. SREG is the register holding
// buf[pos] for this step's static position. DO_SQ==0 only for t==1.
#define ESTEP_CORE(SREG, DO_SQ) do {                                          \
    float x = train[off];                                                     \
    __builtin_prefetch(train + off + 2 * (size_t)B_CONST, 0, 1);              \
    float nl = lev_sm * __fdividef(x, SREG) + oml * lev;                      \
    float ld = __logf(__fdividef(nl, lev));                                   \
    float ns = seas_sm * __fdividef(x, nl) + oms * SREG;                      \
    SREG = ns;                                                                \
    if (valid) {                                                              \
        out_lev[off] = nl;                                                    \
        out_seas[off + (size_t)7 * B_CONST] = ns;                             \
    }                                                                         \
    if (DO_SQ) {                                                              \
        float d = ld - ld_prev;                                               \
        float c = valid ? d * d : 0.0f;                                       \
        c += __shfl_down(c, 16, 32);                                          \
        c += __shfl_down(c, 8, 32);                                           \
        c += __shfl_down(c, 4, 32);                                           \
        c += __shfl_down(c, 2, 32);                                           \
        c += __shfl_down(c, 1, 32);                                           \
        if ((threadIdx.x & 31) == 0) atomicAdd(&lsums[sqidx], c);             \
    }                                                                         \
    ld_prev = ld; lev = nl; off += B_CONST; ++sqidx;                          \
} while (0)

__global__ __launch_bounds__(256) void esrnn_main(
    const float* __restrict__ train,
    const float* __restrict__ init_lev_sms,
    const float* __restrict__ init_seas_sms,
    const float* __restrict__ init_seasonalities,
    float* __restrict__ out_lev,
    float* __restrict__ out_seas,
    float* __restrict__ partials,
    int nblocks)
{
    __shared__ float lsums[NT];
    for (int j = threadIdx.x; j < NT; j += blockDim.x) lsums[j] = 0.0f;
    __syncthreads();

    const int  b     = blockIdx.x * blockDim.x + threadIdx.x;
    const bool valid = (b < B_CONST);
    const int  bb    = valid ? b : (B_CONST - 1);  // safe clamp for loads

    // sigmoids of smoothing params
    const float lev_sm  = 1.0f / (1.0f + __expf(-init_lev_sms[bb]));
    const float seas_sm = 1.0f / (1.0f + __expf(-init_seas_sms[bb]));
    const float oml = 1.0f - lev_sm;
    const float oms = 1.0f - seas_sm;

    // init seasonalities, shape (B, S) row-major
    const float* isrow = init_seasonalities + (size_t)bb * S_CONST;
    float e0 = __expf(isrow[0]);
    float e1 = __expf(isrow[1]);
    float e2 = __expf(isrow[2]);
    float e3 = __expf(isrow[3]);
    float e4 = __expf(isrow[4]);
    float e5 = __expf(isrow[5]);
    float e6 = __expf(isrow[6]);

    if (valid) {  // seas_head rows 0..7 = init_seas.T then seas0 duplicate
        out_seas[(size_t)0 * B_CONST + b] = e0;
        out_seas[(size_t)1 * B_CONST + b] = e1;
        out_seas[(size_t)2 * B_CONST + b] = e2;
        out_seas[(size_t)3 * B_CONST + b] = e3;
        out_seas[(size_t)4 * B_CONST + b] = e4;
        out_seas[(size_t)5 * B_CONST + b] = e5;
        out_seas[(size_t)6 * B_CONST + b] = e6;
        out_seas[(size_t)7 * B_CONST + b] = e0;
    }

    // rotated seasonal buffer: buf[j] = exp_seas[j+1] (j<6), buf[6] = exp_seas[0]
    float s0 = e1, s1 = e2, s2 = e3, s3 = e4, s4 = e5, s5 = e6, s6 = e0;

    float lev = __fdividef(train[bb], e0);   // lev0
    if (valid) out_lev[b] = lev;

    size_t off     = (size_t)B_CONST + bb;   // points at train[t=1][b]
    float  ld_prev = 0.0f;
    int    sqidx   = -1;                     // = t - 2

    // t = 1 (pos 0), no squared-diff contribution yet
    ESTEP_CORE(s0, 0);
    // t = 2 .. 715 : 102 groups of 7, positions 1,2,3,4,5,6,0
    for (int g = 0; g < 102; ++g) {
        ESTEP_CORE(s1, 1);
        ESTEP_CORE(s2, 1);
        ESTEP_CORE(s3, 1);
        ESTEP_CORE(s4, 1);
        ESTEP_CORE(s5, 1);
        ESTEP_CORE(s6, 1);
        ESTEP_CORE(s0, 1);
    }
    // t = 716 .. 719 : positions 1,2,3,4
    ESTEP_CORE(s1, 1);
    ESTEP_CORE(s2, 1);
    ESTEP_CORE(s3, 1);
    ESTEP_CORE(s4, 1);

    __syncthreads();
    // export this block's partial sums, laid out [t][block] for the reducer
    for (int j = threadIdx.x; j < NT; j += blockDim.x)
        partials[(size_t)j * nblocks + blockIdx.x] = lsums[j];
}

// Reduce 196 block-partials per t to the final mean. Expressed as a
// (16 x K) x (K x 16) matmul against an all-ones B operand so it runs on the
// CDNA5 f32 WMMA pipe (V_WMMA_F32_16X16X4_F32): D row-sums == per-t totals.
// One wave (32 lanes, EXEC all-ones) per tile of 16 t-rows.
__global__ __launch_bounds__(32) void esrnn_reduce(
    const float* __restrict__ partials,
    float* __restrict__ out_mean,
    int nblocks)   // must be a multiple of 4 (it is: 196)
{
#if defined(__gfx1250__) && __has_builtin(__builtin_amdgcn_wmma_f32_16x16x4_f32)
    const int  lane  = threadIdx.x;
    const int  mrow  = (lane < 16) ? lane : (lane - 16);
    const int  kb    = (lane < 16) ? 0 : 2;   // K columns this lane holds
    const int  tb    = blockIdx.x * 16;
    const int  trow  = tb + mrow;
    const bool rvalid = (trow < NT);
    const float* rowp = partials + (size_t)(rvalid ? trow : 0) * nblocks;
    const float zmask = rvalid ? 1.0f : 0.0f;

    v2f ones; ones.x = 1.0f; ones.y = 1.0f;   // B = 4x16 all-ones
    v8f acc = {};
    for (int k = 0; k < nblocks; k += 4) {
        v2f a;  // A-layout 16x4 f32: lanes 0-15 hold K=k,k+1; lanes 16-31 K=k+2,k+3
        a.x = rowp[k + kb]     * zmask;
        a.y = rowp[k + kb + 1] * zmask;
        acc = __builtin_amdgcn_wmma_f32_16x16x4_f32(
            false, a, false, ones, (short)0, acc, false, false);
    }
    // D layout: lane 0 has N=0 for M=0..7 in acc[0..7]; lane 16 has M=8..15.
    const float inv = 1.0f / (float)B_CONST;
    if (lane == 0) {
        #pragma unroll
        for (int r = 0; r < 8; ++r) {
            int t = tb + r;
            if (t < NT) out_mean[t] = acc[r] * inv;
        }
    } else if (lane == 16) {
        #pragma unroll
        for (int r = 0; r < 8; ++r) {
            int t = tb + 8 + r;
            if (t < NT) out_mean[t] = acc[r] * inv;
        }
    }
#else
    int j = blockIdx.x * blockDim.x + threadIdx.x;
    if (j >= NT) return;
    const float* p = partials + (size_t)j * nblocks;
    float s = 0.0f;
    for (int k = 0; k < nblocks; ++k) s += p[k];
    out_mean[j] = s * (1.0f / (float)B_CONST);
#endif
}

extern "C" void kernel_launch(void* const* d_in, const int* in_sizes, int n_in,
                              void* d_out, int out_size, void* d_ws, size_t ws_size,
                              hipStream_t stream) {
    const float* train = (const float*)d_in[0];   // (720, 50000)
    const float* ilev  = (const float*)d_in[1];   // (50000,)
    const float* isms  = (const float*)d_in[2];   // (50000,)
    const float* iseas = (const float*)d_in[3];   // (50000, 7)

    float* out      = (float*)d_out;
    float* out_lev  = out;                                   // 720*B
    float* out_seas = out + (size_t)T_CONST * B_CONST;       // 727*B
    float* out_mean = out_seas + (size_t)(S_CONST + 1 + T_CONST - 1) * B_CONST; // 718

    float* partials = (float*)d_ws;  // [NT][nblocks] floats (~563 KB)

    const int nblocks = (B_CONST + 255) / 256;   // 196, multiple of 4
    esrnn_main<<<nblocks, 256, 0, stream>>>(train, ilev, isms, iseas,
                                            out_lev, out_seas, partials, nblocks);

    const int ntiles = (NT + 15) / 16;           // 45 tiles of 16 t-rows
    esrnn_reduce<<<ntiles, 32, 0, stream>>>(partials, out_mean, nblocks);
}